// SurfaceMap_34772055228972
// MI455X (gfx1250) — compile-verified
//
#include <hip/hip_runtime.h>

typedef __attribute__((ext_vector_type(2))) float v2f;
typedef __attribute__((ext_vector_type(8))) float v8f;

#define NPTS            524288
#define NGAUSS          64
#define NMONOM          16
#define THREADS         256
#define WAVES_PER_BLOCK (THREADS / 32)
#define TILES_PER_WAVE  4
#define NTILES          (NPTS / 16)                                   // 32768
#define NBLOCKS         (NTILES / (TILES_PER_WAVE * WAVES_PER_BLOCK)) // 1024

// ws float layout:
//   [g*8 + 0..5] : log2(e)-scaled quadratic-form coefficients for gaussian g,
//                  with -log2(g_const) folded into the constant term, so that
//                  2^(-(feat . cf)) == g_const * exp(-zeta)
//   [512 + p*4+q]: summed polynomial coefficient for x^p * y^q

__global__ void sm_prep(const float* __restrict__ means,
                        const float* __restrict__ covs,
                        const float* __restrict__ g_consts,
                        const float* __restrict__ p_consts,
                        const int*   __restrict__ pows,
                        float* __restrict__ ws) {
  int t = threadIdx.x;
  if (t < 16) ws[512 + t] = 0.0f;           // ws is poisoned 0xAA; must zero the table
  __syncthreads();
  if (t < NGAUSS) {
    const float L2E = 1.4426950408889634f;
    float c00 = covs[t*4+0], c01 = covs[t*4+1], c10 = covs[t*4+2], c11 = covs[t*4+3];
    float mx  = means[t*2+0], my = means[t*2+1];
    float b   = c01 + c10;
    float* cf = ws + t*8;
    cf[0] = L2E * c00;                                  // * x^2
    cf[1] = L2E * b;                                    // * x*y
    cf[2] = L2E * c11;                                  // * y^2
    cf[3] = L2E * (-2.0f*c00*mx - b*my);                // * x
    cf[4] = L2E * (-2.0f*c11*my - b*mx);                // * y
    cf[5] = L2E * (c00*mx*mx + b*mx*my + c11*my*my)     // * 1, with gc folded in
            - log2f(g_consts[t]);
    cf[6] = 0.0f; cf[7] = 0.0f;
  }
  if (t < NMONOM) {
    int p = pows[t*2+0], q = pows[t*2+1];               // powers in [0,4)
    atomicAdd(&ws[512 + p*4 + q], p_consts[t]);         // duplicates (p,q) may occur
  }
}

// One 16-point tile vs all 64 gaussians.
// A (SRC0) = gaussian coefficients (M=16 gaussians/block, K=6 in two K=4 tiles);
// B (SRC1) = point features (K=6, N=16 points).
// D layout: lane l holds point N=(l&15); vgpr r holds gaussian M = r + 8*(l>=16),
// so the gaussian reduction is 8 in-lane adds per block; halves merged by caller.
__device__ __forceinline__ float tile_gauss(const v2f A1[4], const v2f A2[4],
                                            v2f Bf1, v2f Bf2) {
  float s = 0.0f;
#pragma unroll
  for (int b = 0; b < 4; ++b) {
    v8f acc = {0.f, 0.f, 0.f, 0.f, 0.f, 0.f, 0.f, 0.f};
    acc = __builtin_amdgcn_wmma_f32_16x16x4_f32(false, A2[b], false, Bf2,
                                                (short)0, acc, false, false);
    acc = __builtin_amdgcn_wmma_f32_16x16x4_f32(false, A1[b], false, Bf1,
                                                (short)0, acc, false, false);
    // acc = log2e*zeta - log2(gc)  ->  2^(-acc) = gc * exp(-zeta)
    float e0 = __builtin_amdgcn_exp2f(-acc[0]);
    float e1 = __builtin_amdgcn_exp2f(-acc[1]);
    float e2 = __builtin_amdgcn_exp2f(-acc[2]);
    float e3 = __builtin_amdgcn_exp2f(-acc[3]);
    float e4 = __builtin_amdgcn_exp2f(-acc[4]);
    float e5 = __builtin_amdgcn_exp2f(-acc[5]);
    float e6 = __builtin_amdgcn_exp2f(-acc[6]);
    float e7 = __builtin_amdgcn_exp2f(-acc[7]);
    s += ((e0 + e1) + (e2 + e3)) + ((e4 + e5) + (e6 + e7));
  }
  return s;
}

__global__ void __launch_bounds__(THREADS)
sm_main(const float* __restrict__ points,
        const float* __restrict__ ws,
        float* __restrict__ out) {
  const int  lane  = threadIdx.x & 31;
  const bool hiH   = lane >= 16;                  // upper half-wave holds K=2,3 / M=8..15
  const int  l15   = lane & 15;
  const int  gwave = blockIdx.x * WAVES_PER_BLOCK + (threadIdx.x >> 5);

  // A matrices: 4 gaussian blocks x (two K=4 tiles).
  // A layout (32-bit, 16x4): lanes 0-15: vgpr0=K0,vgpr1=K1 (M=lane);
  //                          lanes 16-31: vgpr0=K2,vgpr1=K3 (M=lane-16).
  v2f A1[4], A2[4];
#pragma unroll
  for (int b = 0; b < 4; ++b) {
    const float* cf = ws + (b*16 + l15) * 8;
    float a0 = cf[0], a1 = cf[1], a2 = cf[2], a3 = cf[3], a4 = cf[4], a5 = cf[5];
    A1[b] = hiH ? v2f{a2, a3} : v2f{a0, a1};
    A2[b] = hiH ? v2f{0.0f, 0.0f} : v2f{a4, a5};  // K6,K7 zero-padded
  }

  // polynomial coefficient table: poly = sum_{p,q} tb[p*4+q] * x^p * y^q
  float tb[16];
#pragma unroll
  for (int i = 0; i < 16; ++i) tb[i] = ws[512 + i];

#pragma unroll
  for (int t = 0; t < TILES_PER_WAVE; t += 2) {   // two 16-point tiles per iteration
    const int pbase = (gwave * TILES_PER_WAVE + t) * 16;

    const float xA = points[(pbase + l15)*2 + 0];
    const float yA = points[(pbase + l15)*2 + 1];
    const float xB = points[(pbase + 16 + l15)*2 + 0];
    const float yB = points[(pbase + 16 + l15)*2 + 1];

    // B feature tiles (4x16): lanes 0-15: {K0,K1}={x2,xy}; lanes 16-31: {K2,K3}={y2,x}
    const v2f BA1 = hiH ? v2f{yA*yA, xA}      : v2f{xA*xA, xA*yA};
    const v2f BA2 = hiH ? v2f{0.0f, 0.0f}     : v2f{yA, 1.0f};
    const v2f BB1 = hiH ? v2f{yB*yB, xB}      : v2f{xB*xB, xB*yB};
    const v2f BB2 = hiH ? v2f{0.0f, 0.0f}     : v2f{yB, 1.0f};

    float sA = tile_gauss(A1, A2, BA1, BA2);    // lane sum over its 8 gauss rows x 4 blocks
    float sB = tile_gauss(A1, A2, BB1, BB2);

    // merge the two half-wave gaussian subsets: one xor-16 shuffle per tile
    sA += __shfl_xor(sA, 16, 32);
    sB += __shfl_xor(sB, 16, 32);

    // branchless finish: low half finalizes tile A (points pbase+0..15),
    // high half finalizes tile B (points pbase+16..31).
    const float px = hiH ? xB : xA;
    const float py = hiH ? yB : yA;
    const float gs = hiH ? sB : sA;

    const float px2 = px*px, py2 = py*py;
    const float px3 = px2*px, py3 = py2*py;
    float i0 = tb[0]  + tb[1]*py  + tb[2]*py2  + tb[3]*py3;
    float i1 = tb[4]  + tb[5]*py  + tb[6]*py2  + tb[7]*py3;
    float i2 = tb[8]  + tb[9]*py  + tb[10]*py2 + tb[11]*py3;
    float i3 = tb[12] + tb[13]*py + tb[14]*py2 + tb[15]*py3;
    float poly = i0 + px*i1 + px2*i2 + px3*i3;

    out[pbase + lane] = gs + poly;              // full 32-lane coalesced store
  }
}

extern "C" void kernel_launch(void* const* d_in, const int* in_sizes, int n_in,
                              void* d_out, int out_size, void* d_ws, size_t ws_size,
                              hipStream_t stream) {
  (void)in_sizes; (void)n_in; (void)out_size; (void)ws_size;
  const float* points   = (const float*)d_in[0];
  const float* means    = (const float*)d_in[1];
  const float* covs     = (const float*)d_in[2];
  const float* g_consts = (const float*)d_in[3];
  const float* p_consts = (const float*)d_in[4];
  const int*   pows     = (const int*)d_in[5];
  float* out = (float*)d_out;
  float* ws  = (float*)d_ws;

  sm_prep<<<1, 64, 0, stream>>>(means, covs, g_consts, p_consts, pows, ws);
  sm_main<<<NBLOCKS, THREADS, 0, stream>>>(points, ws, out);
}